// KMeansQuantizer_9706626089636
// MI455X (gfx1250) — compile-verified
//
#include <hip/hip_runtime.h>
#include <math.h>

typedef __attribute__((ext_vector_type(2))) float v2f;
typedef __attribute__((ext_vector_type(8))) float v8f;

#define DIM   64
#define KCB   4096
#define NTOK  65536
#define BN    64                 // e columns staged in LDS per iteration
#define PPITCH 160               // dwords per d-pair row: 160 % 64 == 32 -> conflict-free b64 reads
#define PBUF  (32 * PPITCH)      // floats per buffer (32 d-pairs)

// ---------------------------------------------------------------------------
// 0) zero scatter accumulators (S, counts, lossacc, nval are contiguous in ws)
__global__ void zero_kernel(float* p, int n) {
    int i = blockIdx.x * 256 + threadIdx.x;
    if (i < n) p[i] = 0.0f;
}

// ---------------------------------------------------------------------------
// 1) cn[k] = sum_d e[d][k]^2   (coalesced: consecutive k across lanes)
__global__ void colnorm_kernel(const float* __restrict__ e, float* __restrict__ cn) {
    int k = blockIdx.x * 256 + threadIdx.x;
    float s = 0.0f;
    #pragma unroll
    for (int d = 0; d < DIM; ++d) {
        float v = e[(size_t)d * KCB + k];
        s += v * v;
    }
    cn[k] = s;
}

// ---------------------------------------------------------------------------
// 2) distance GEMM + argmin via V_WMMA_F32_16X16X4_F32.
//    dist[m][k] = cn[k] - 2 * dot(x[m], e[:,k])  (||x||^2 dropped: constant per row)
//    A = -2*x: TWO 16-row tiles per wave (M=32) so every B fetch feeds 2 WMMAs.
//    e tiles are software-pipelined: global b128 loads for tile i+1 issue before
//    the 128-WMMA compute of tile i, get register-transposed into a k-pair-
//    interleaved LDS layout (pairs {e[r][n],e[r+1][n]} contiguous, r always even)
//    so each B operand is ONE aligned ds_load_b64 straight into the WMMA pair.
__global__ __launch_bounds__(256) void dist_argmin_kernel(
    const float* __restrict__ x, const float* __restrict__ e,
    const float* __restrict__ cn, int* __restrict__ enc)
{
    __shared__ __align__(16) float lds[2 * PBUF];   // 2 x 20 KB, double-buffered

    const int t    = threadIdx.x;
    const int wave = t >> 5;
    const int lane = t & 31;
    const int half = lane >> 4;          // 0: lanes 0-15, 1: lanes 16-31
    const int nl   = lane & 15;
    const int m0   = blockIdx.x * 256 + wave * 32;

    // --- staged fill: thread t owns d-pair p = t>>3, col chunk n0 = (t&7)*8 ---
    const int p  = t >> 3;               // 0..31  (d rows 2p, 2p+1)
    const int n0 = (t & 7) * 8;          // 0,8,...,56
    float4 g0, g1, g2, g3;               // prefetch regs: rows 2p/2p+1, col halves

    auto load_tile = [&](int k0) {
        const float* base = e + (size_t)(2 * p) * KCB + k0 + n0;
        g0 = *(const float4*)(base);
        g1 = *(const float4*)(base + 4);
        g2 = *(const float4*)(base + KCB);
        g3 = *(const float4*)(base + KCB + 4);
    };
    auto store_tile = [&](int buf) {
        float* lb = &lds[buf * PBUF + p * PPITCH + 2 * n0];
        float4 s0 = {g0.x, g2.x, g0.y, g2.y};   // cols n0, n0+1
        float4 s1 = {g0.z, g2.z, g0.w, g2.w};   // cols n0+2, n0+3
        float4 s2 = {g1.x, g3.x, g1.y, g3.y};   // cols n0+4, n0+5
        float4 s3 = {g1.z, g3.z, g1.w, g3.w};   // cols n0+6, n0+7
        *(float4*)(lb + 0)  = s0;
        *(float4*)(lb + 4)  = s1;
        *(float4*)(lb + 8)  = s2;
        *(float4*)(lb + 12) = s3;
    };

    load_tile(0);

    // Preload A = -2*x for both 16-row tiles. f32 16x4 A layout per step s:
    // lanes 0-15 hold K=0,1 (v.x,v.y), lanes 16-31 hold K=2,3 (d = 4s+2*half+j).
    v2f a0[16], a1[16];
    {
        const float* xr0 = x + (size_t)(m0 + nl) * DIM;
        const float* xr1 = x + (size_t)(m0 + 16 + nl) * DIM;
        #pragma unroll
        for (int s = 0; s < 16; ++s) {
            float2 u = *(const float2*)(xr0 + 4 * s + 2 * half);
            float2 v = *(const float2*)(xr1 + 4 * s + 2 * half);
            a0[s].x = -2.0f * u.x;  a0[s].y = -2.0f * u.y;
            a1[s].x = -2.0f * v.x;  a1[s].y = -2.0f * v.y;
        }
    }

    store_tile(0);
    __syncthreads();

    float bv0[8], bv1[8];
    int   bi0[8], bi1[8];
    #pragma unroll
    for (int j = 0; j < 8; ++j) {
        bv0[j] = 3.4e38f; bi0[j] = 0;
        bv1[j] = 3.4e38f; bi1[j] = 0;
    }

    for (int it = 0; it < KCB / BN; ++it) {
        const int k0  = it * BN;
        const int buf = it & 1;
        const float* lbase = &lds[buf * PBUF];

        if (it + 1 < KCB / BN) load_tile(k0 + BN);   // issue early: hides under WMMAs

        float cnv[4];
        #pragma unroll
        for (int sub = 0; sub < 4; ++sub) cnv[sub] = cn[k0 + sub * 16 + nl];

        #pragma unroll
        for (int sp = 0; sp < 2; ++sp) {
            const int nA = sp * 32 + nl;             // subtile 2*sp
            const int nB = nA + 16;                  // subtile 2*sp+1

            v8f c00, c01, c10, c11;                  // [mtile][ntile]
            #pragma unroll
            for (int j = 0; j < 8; ++j) {
                c00[j] = cnv[2 * sp];     c10[j] = cnv[2 * sp];
                c01[j] = cnv[2 * sp + 1]; c11[j] = cnv[2 * sp + 1];
            }

            // base for half-dependent pair row: (r>>1) = 2s + half
            const float* bpA = lbase + half * PPITCH + 2 * nA;
            const float* bpB = lbase + half * PPITCH + 2 * nB;
            #pragma unroll
            for (int s = 0; s < 16; ++s) {
                v2f b0 = *(const v2f*)(bpA + 2 * s * PPITCH);   // {e[r][nA], e[r+1][nA]}
                v2f b1 = *(const v2f*)(bpB + 2 * s * PPITCH);
                c00 = __builtin_amdgcn_wmma_f32_16x16x4_f32(
                          false, a0[s], false, b0, (short)0, c00, false, false);
                c10 = __builtin_amdgcn_wmma_f32_16x16x4_f32(
                          false, a1[s], false, b0, (short)0, c10, false, false);
                c01 = __builtin_amdgcn_wmma_f32_16x16x4_f32(
                          false, a0[s], false, b1, (short)0, c01, false, false);
                c11 = __builtin_amdgcn_wmma_f32_16x16x4_f32(
                          false, a1[s], false, b1, (short)0, c11, false, false);
            }

            const int kA = k0 + nA, kB = k0 + nB;
            #pragma unroll
            for (int j = 0; j < 8; ++j) {
                if (c00[j] < bv0[j]) { bv0[j] = c00[j]; bi0[j] = kA; }
                if (c01[j] < bv0[j]) { bv0[j] = c01[j]; bi0[j] = kB; }
                if (c10[j] < bv1[j]) { bv1[j] = c10[j]; bi1[j] = kA; }
                if (c11[j] < bv1[j]) { bv1[j] = c11[j]; bi1[j] = kB; }
            }
        }

        if (it + 1 < KCB / BN) store_tile(buf ^ 1);  // waits its loads via data dep
        __syncthreads();
    }

    // Reduce across the 16 lanes of each half (C rows: m = j + 8*half).
    // Tie-break on lower index to match jnp.argmin.
    #pragma unroll
    for (int off = 1; off < 16; off <<= 1) {
        #pragma unroll
        for (int j = 0; j < 8; ++j) {
            float ov0 = __shfl_xor(bv0[j], off, 16);
            int   oi0 = __shfl_xor(bi0[j], off, 16);
            if (ov0 < bv0[j] || (ov0 == bv0[j] && oi0 < bi0[j])) { bv0[j] = ov0; bi0[j] = oi0; }
            float ov1 = __shfl_xor(bv1[j], off, 16);
            int   oi1 = __shfl_xor(bi1[j], off, 16);
            if (ov1 < bv1[j] || (ov1 == bv1[j] && oi1 < bi1[j])) { bv1[j] = ov1; bi1[j] = oi1; }
        }
    }
    if (nl == 0) {
        #pragma unroll
        for (int j = 0; j < 8; ++j) {
            enc[m0 + 8 * half + j]      = bi0[j];
            enc[m0 + 16 + 8 * half + j] = bi1[j];
        }
    }
}

// ---------------------------------------------------------------------------
// 3) gather quantized, straight-through output, loss partial, scatter counts/sums.
//    16 tokens per 1024-thread block; 64 threads (one per d) per token.
__global__ __launch_bounds__(1024) void gather_scatter_kernel(
    const float* __restrict__ x, const float* __restrict__ e,
    const int* __restrict__ enc, float* __restrict__ out_q,
    float* __restrict__ counts, float* __restrict__ S,
    float* __restrict__ lossacc)
{
    __shared__ float red[1024];
    const int t   = threadIdx.x;
    const int tok = blockIdx.x * 16 + (t >> 6);
    const int d   = t & 63;
    const int idx = enc[tok];

    const float xv = x[(size_t)tok * DIM + d];
    const float qv = e[(size_t)d * KCB + idx];       // e column gather (L2-resident)
    out_q[(size_t)tok * DIM + d] = xv + (qv - xv);   // straight-through

    const float diff = qv - xv;
    red[t] = diff * diff;

    atomicAdd(&S[(size_t)d * KCB + idx], xv);        // sums^T scatter [D,K]
    if (d == 0) atomicAdd(&counts[idx], 1.0f);

    __syncthreads();
    #pragma unroll
    for (int off = 512; off > 0; off >>= 1) {
        if (t < off) red[t] += red[t + off];
        __syncthreads();
    }
    if (t == 0) atomicAdd(lossacc, red[0]);
}

// ---------------------------------------------------------------------------
// 4) EMA cluster sizes + scalars (single block).
__global__ __launch_bounds__(1024) void finalize1_kernel(
    const float* __restrict__ counts, const float* __restrict__ cs,
    float* __restrict__ out_cs, float* __restrict__ out_loss,
    float* __restrict__ out_ppl, const float* __restrict__ lossacc,
    float* __restrict__ nval)
{
    __shared__ float s_n[1024];
    __shared__ float s_p[1024];
    const int t = threadIdx.x;
    float np = 0.0f, pp = 0.0f;
    for (int i = 0; i < KCB; i += 1024) {
        int k = i + t;
        float c   = counts[k];
        float ncs = 0.1f * c + 0.9f * cs[k];
        out_cs[k] = ncs;
        np += ncs;
        float avg = c * (1.0f / (float)NTOK);
        pp += avg * logf(avg + 1e-20f);
    }
    s_n[t] = np; s_p[t] = pp;
    __syncthreads();
    for (int off = 512; off > 0; off >>= 1) {
        if (t < off) { s_n[t] += s_n[t + off]; s_p[t] += s_p[t + off]; }
        __syncthreads();
    }
    if (t == 0) {
        nval[0]     = s_n[0];
        out_ppl[0]  = expf(-s_p[0]);
        out_loss[0] = 0.25f * lossacc[0] * (1.0f / ((float)NTOK * (float)DIM));
    }
}

// ---------------------------------------------------------------------------
// 5) new_un = 0.1*S + 0.9*un ;  new_e = new_un / stable_cs
__global__ void finalize2_kernel(
    const float* __restrict__ S, const float* __restrict__ un,
    const float* __restrict__ new_cs, const float* __restrict__ nval,
    float* __restrict__ out_un, float* __restrict__ out_e)
{
    const int i = blockIdx.x * 256 + threadIdx.x;   // 0 .. D*K-1
    const int k = i & (KCB - 1);
    const float nu = 0.1f * S[i] + 0.9f * un[i];
    out_un[i] = nu;
    const float n = nval[0];
    const float stable = (new_cs[k] + 1e-20f) / (n + (float)KCB * 1e-20f) * n;
    out_e[i] = nu / stable;
}

// ---------------------------------------------------------------------------
extern "C" void kernel_launch(void* const* d_in, const int* in_sizes, int n_in,
                              void* d_out, int out_size, void* d_ws, size_t ws_size,
                              hipStream_t stream) {
    const float* x  = (const float*)d_in[0];   // [N, D]
    const float* e  = (const float*)d_in[1];   // [D, K]
    const float* cs = (const float*)d_in[2];   // [K]
    const float* un = (const float*)d_in[3];   // [D, K]

    float* out = (float*)d_out;
    float* ws  = (float*)d_ws;

    // workspace layout
    float* cn      = ws;                        // K
    float* S       = cn + KCB;                  // D*K   (scatter sums^T)
    float* counts  = S + (size_t)DIM * KCB;     // K
    float* lossacc = counts + KCB;              // 1
    float* nval    = lossacc + 1;               // 1
    int*   enc     = (int*)(nval + 1);          // N

    // output layout (reference tuple, flattened in order)
    float* out_q    = out;                                  // N*D
    float* out_loss = out + (size_t)NTOK * DIM;             // 1
    float* out_ppl  = out_loss + 1;                         // 1
    float* out_e    = out_ppl + 1;                          // D*K
    float* out_cs   = out_e + (size_t)DIM * KCB;            // K
    float* out_un   = out_cs + KCB;                         // D*K

    const int nzero = DIM * KCB + KCB + 2;  // S, counts, lossacc, nval
    zero_kernel<<<(nzero + 255) / 256, 256, 0, stream>>>(S, nzero);
    colnorm_kernel<<<KCB / 256, 256, 0, stream>>>(e, cn);
    dist_argmin_kernel<<<NTOK / 256, 256, 0, stream>>>(x, e, cn, enc);
    gather_scatter_kernel<<<NTOK / 16, 1024, 0, stream>>>(x, e, enc, out_q, counts, S, lossacc);
    finalize1_kernel<<<1, 1024, 0, stream>>>(counts, cs, out_cs, out_loss, out_ppl, lossacc, nval);
    finalize2_kernel<<<(DIM * KCB) / 256, 256, 0, stream>>>(S, un, out_cs, nval, out_un, out_e);
}